// MambaFusionV2_82944408420944
// MI455X (gfx1250) — compile-verified
//
#include <hip/hip_runtime.h>
#include <math.h>

// MambaFusionV2 forward for gfx1250 (MI455X, wave32).
// All GEMMs via V_WMMA_F32_16X16X4_F32 (fp32 in/out, full precision vs fp32 reference).

typedef float v2f __attribute__((ext_vector_type(2)));
typedef float v8f __attribute__((ext_vector_type(8)));

#define EPSV 1e-5f

#define WMMA_F32(Afr, Bfr, Cacc) \
    __builtin_amdgcn_wmma_f32_16x16x4_f32(false, (Afr), false, (Bfr), (short)0, (Cacc), false, false)

// ---------------- block reduction (blockDim.x == 256) ----------------
__device__ __forceinline__ void blockReduce2(float& s0, float& s1) {
    __shared__ float r0[256];
    __shared__ float r1[256];
    int t = threadIdx.x;
    r0[t] = s0; r1[t] = s1;
    __syncthreads();
    for (int off = 128; off > 0; off >>= 1) {
        if (t < off) { r0[t] += r0[t + off]; r1[t] += r1[t + off]; }
        __syncthreads();
    }
    float a = r0[0], b = r1[0];
    __syncthreads();
    s0 = a; s1 = b;
}

// ---------------- WMMA f32 GEMM:  C[M,N] = A[M,K] * W[N,K]^T (+bias, +softplus) -----
// amode 0: A element (m,k) at A[m*lda + k]
// amode 1: A element (m,k) at A[(m/aL)*K*aL + k*aL + (m%aL)]   (NCHW conv1x1 input)
// cmode 0: C[m*N + n]
// cmode 1: C[(m/cL)*cCtot*cL + n*cL + (m%cL)]                  (NCHW output)
// act 1: softplus(v)
__device__ __forceinline__ void gemm_store_tile(v8f acc, int m0, int hi, int ncol,
                                                float* __restrict__ C,
                                                const float* __restrict__ bias,
                                                int N, int act, int cmode, int cL, int cCtot) {
    float bia = bias ? bias[ncol] : 0.f;
#pragma unroll
    for (int r = 0; r < 8; ++r) {
        int mm = m0 + hi * 8 + r;
        float v = acc[r] + bia;
        if (act == 1) v = (v > 20.f) ? v : log1pf(expf(v));
        if (cmode == 0) {
            C[(size_t)mm * (size_t)N + (size_t)ncol] = v;
        } else {
            int bi = mm / cL; int px = mm - bi * cL;
            C[((size_t)bi * (size_t)cCtot + (size_t)ncol) * (size_t)cL + (size_t)px] = v;
        }
    }
}

__global__ void wmma_gemm_f32(const float* __restrict__ A, const float* __restrict__ W,
                              float* __restrict__ C, const float* __restrict__ bias,
                              int M, int N, int K, int lda,
                              int amode, int aL,
                              int cmode, int cL, int cCtot, int act) {
    int wave = threadIdx.x >> 5;
    int lane = threadIdx.x & 31;
    int m0 = blockIdx.y * 128 + wave * 16;
    if (m0 >= M) return;                       // uniform per wave
    int n0 = blockIdx.x << 6;
    int row = lane & 15;
    int hi = lane >> 4;
    int koff = hi << 1;
    int m = m0 + row;

    size_t abase, astr;
    if (amode == 0) { abase = (size_t)m * (size_t)lda; astr = 1; }
    else {
        int bi = m / aL; int px = m - bi * aL;
        abase = (size_t)bi * (size_t)K * (size_t)aL + (size_t)px;
        astr = (size_t)aL;
    }
    const float* Arow = A + abase + (size_t)koff * astr;

    int nt = (N - n0) >> 4; if (nt > 4) nt = 4;
    const float* w0 = W + (size_t)(n0 +  0 + row) * (size_t)K + koff;
    const float* w1 = W + (size_t)(n0 + 16 + row) * (size_t)K + koff;
    const float* w2 = W + (size_t)(n0 + 32 + row) * (size_t)K + koff;
    const float* w3 = W + (size_t)(n0 + 48 + row) * (size_t)K + koff;

    v8f acc0 = {0.f,0.f,0.f,0.f,0.f,0.f,0.f,0.f};
    v8f acc1 = acc0, acc2 = acc0, acc3 = acc0;

    if (nt == 4 && amode == 0 && (K & 15) == 0) {
        // Hot path: all-full tiles, token-major A, K % 16 == 0.
        // Double-buffered software pipeline over 16-wide K blocks:
        // loads for block i+1 are issued before the 16 WMMAs consuming block i.
        v2f a0  = *(const v2f*)(Arow),      a1  = *(const v2f*)(Arow + 4);
        v2f a2  = *(const v2f*)(Arow + 8),  a3  = *(const v2f*)(Arow + 12);
        v2f b00 = *(const v2f*)(w0),        b10 = *(const v2f*)(w1);
        v2f b20 = *(const v2f*)(w2),        b30 = *(const v2f*)(w3);
        v2f b01 = *(const v2f*)(w0 + 4),    b11 = *(const v2f*)(w1 + 4);
        v2f b21 = *(const v2f*)(w2 + 4),    b31 = *(const v2f*)(w3 + 4);
        v2f b02 = *(const v2f*)(w0 + 8),    b12 = *(const v2f*)(w1 + 8);
        v2f b22 = *(const v2f*)(w2 + 8),    b32 = *(const v2f*)(w3 + 8);
        v2f b03 = *(const v2f*)(w0 + 12),   b13 = *(const v2f*)(w1 + 12);
        v2f b23 = *(const v2f*)(w2 + 12),   b33 = *(const v2f*)(w3 + 12);

#pragma unroll 1
        for (int k = 16; k < K; k += 16) {
            // issue next block's loads
            v2f na0  = *(const v2f*)(Arow + k),      na1  = *(const v2f*)(Arow + k + 4);
            v2f na2  = *(const v2f*)(Arow + k + 8),  na3  = *(const v2f*)(Arow + k + 12);
            v2f nb00 = *(const v2f*)(w0 + k),        nb10 = *(const v2f*)(w1 + k);
            v2f nb20 = *(const v2f*)(w2 + k),        nb30 = *(const v2f*)(w3 + k);
            v2f nb01 = *(const v2f*)(w0 + k + 4),    nb11 = *(const v2f*)(w1 + k + 4);
            v2f nb21 = *(const v2f*)(w2 + k + 4),    nb31 = *(const v2f*)(w3 + k + 4);
            v2f nb02 = *(const v2f*)(w0 + k + 8),    nb12 = *(const v2f*)(w1 + k + 8);
            v2f nb22 = *(const v2f*)(w2 + k + 8),    nb32 = *(const v2f*)(w3 + k + 8);
            v2f nb03 = *(const v2f*)(w0 + k + 12),   nb13 = *(const v2f*)(w1 + k + 12);
            v2f nb23 = *(const v2f*)(w2 + k + 12),   nb33 = *(const v2f*)(w3 + k + 12);

            // compute on current block
            acc0 = WMMA_F32(a0, b00, acc0);
            acc1 = WMMA_F32(a0, b10, acc1);
            acc2 = WMMA_F32(a0, b20, acc2);
            acc3 = WMMA_F32(a0, b30, acc3);
            acc0 = WMMA_F32(a1, b01, acc0);
            acc1 = WMMA_F32(a1, b11, acc1);
            acc2 = WMMA_F32(a1, b21, acc2);
            acc3 = WMMA_F32(a1, b31, acc3);
            acc0 = WMMA_F32(a2, b02, acc0);
            acc1 = WMMA_F32(a2, b12, acc1);
            acc2 = WMMA_F32(a2, b22, acc2);
            acc3 = WMMA_F32(a2, b32, acc3);
            acc0 = WMMA_F32(a3, b03, acc0);
            acc1 = WMMA_F32(a3, b13, acc1);
            acc2 = WMMA_F32(a3, b23, acc2);
            acc3 = WMMA_F32(a3, b33, acc3);

            // rotate buffers
            a0 = na0;  a1 = na1;  a2 = na2;  a3 = na3;
            b00 = nb00; b10 = nb10; b20 = nb20; b30 = nb30;
            b01 = nb01; b11 = nb11; b21 = nb21; b31 = nb31;
            b02 = nb02; b12 = nb12; b22 = nb22; b32 = nb32;
            b03 = nb03; b13 = nb13; b23 = nb23; b33 = nb33;
        }
        // drain last block
        acc0 = WMMA_F32(a0, b00, acc0);
        acc1 = WMMA_F32(a0, b10, acc1);
        acc2 = WMMA_F32(a0, b20, acc2);
        acc3 = WMMA_F32(a0, b30, acc3);
        acc0 = WMMA_F32(a1, b01, acc0);
        acc1 = WMMA_F32(a1, b11, acc1);
        acc2 = WMMA_F32(a1, b21, acc2);
        acc3 = WMMA_F32(a1, b31, acc3);
        acc0 = WMMA_F32(a2, b02, acc0);
        acc1 = WMMA_F32(a2, b12, acc1);
        acc2 = WMMA_F32(a2, b22, acc2);
        acc3 = WMMA_F32(a2, b32, acc3);
        acc0 = WMMA_F32(a3, b03, acc0);
        acc1 = WMMA_F32(a3, b13, acc1);
        acc2 = WMMA_F32(a3, b23, acc2);
        acc3 = WMMA_F32(a3, b33, acc3);
    } else if (nt == 4) {
        // Full tiles, strided (channel-major) A; branch-free, 2 K-steps unrolled.
        for (int k = 0; k < K; k += 8) {
            v2f a0, a1;
            a0.x = Arow[(size_t)k * astr];
            a0.y = Arow[(size_t)k * astr + astr];
            a1.x = Arow[(size_t)(k + 4) * astr];
            a1.y = Arow[(size_t)(k + 4) * astr + astr];
            v2f b00 = *(const v2f*)(w0 + k),     b10 = *(const v2f*)(w1 + k);
            v2f b20 = *(const v2f*)(w2 + k),     b30 = *(const v2f*)(w3 + k);
            v2f b01 = *(const v2f*)(w0 + k + 4), b11 = *(const v2f*)(w1 + k + 4);
            v2f b21 = *(const v2f*)(w2 + k + 4), b31 = *(const v2f*)(w3 + k + 4);
            acc0 = WMMA_F32(a0, b00, acc0);
            acc1 = WMMA_F32(a0, b10, acc1);
            acc2 = WMMA_F32(a0, b20, acc2);
            acc3 = WMMA_F32(a0, b30, acc3);
            acc0 = WMMA_F32(a1, b01, acc0);
            acc1 = WMMA_F32(a1, b11, acc1);
            acc2 = WMMA_F32(a1, b21, acc2);
            acc3 = WMMA_F32(a1, b31, acc3);
        }
    } else {
        // Generic guarded fallback (partial N span).
        for (int k = 0; k < K; k += 4) {
            v2f a;
            a.x = Arow[(size_t)k * astr];
            a.y = Arow[(size_t)k * astr + astr];
            {
                v2f b; b.x = w0[k]; b.y = w0[k + 1];
                acc0 = WMMA_F32(a, b, acc0);
            }
            if (nt > 1) {
                v2f b; b.x = w1[k]; b.y = w1[k + 1];
                acc1 = WMMA_F32(a, b, acc1);
            }
            if (nt > 2) {
                v2f b; b.x = w2[k]; b.y = w2[k + 1];
                acc2 = WMMA_F32(a, b, acc2);
            }
            if (nt > 3) {
                v2f b; b.x = w3[k]; b.y = w3[k + 1];
                acc3 = WMMA_F32(a, b, acc3);
            }
        }
    }

    gemm_store_tile(acc0, m0, hi, n0 + 0  + row, C, bias, N, act, cmode, cL, cCtot);
    if (nt > 1) gemm_store_tile(acc1, m0, hi, n0 + 16 + row, C, bias, N, act, cmode, cL, cCtot);
    if (nt > 2) gemm_store_tile(acc2, m0, hi, n0 + 32 + row, C, bias, N, act, cmode, cL, cCtot);
    if (nt > 3) gemm_store_tile(acc3, m0, hi, n0 + 48 + row, C, bias, N, act, cmode, cL, cCtot);
}

// ---------------- GroupNorm(32) + SiLU, layout (B, Ctot, Np), slice at c0 ----------
__global__ void gn_silu_kernel(const float* __restrict__ in, float* __restrict__ out,
                               const float* __restrict__ gamma, const float* __restrict__ beta,
                               int C, int Np, int ctotIn, int c0In, int ctotOut, int c0Out) {
    int g = blockIdx.x; int b = blockIdx.y;
    int cpg = C >> 5;
    int nelem = cpg * Np;
    size_t ibase = ((size_t)b * ctotIn + c0In + g * cpg) * (size_t)Np;
    size_t obase = ((size_t)b * ctotOut + c0Out + g * cpg) * (size_t)Np;
    float s = 0.f, ss = 0.f;
    for (int e = threadIdx.x; e < nelem; e += 256) {
        float v = in[ibase + e]; s += v; ss += v * v;
    }
    blockReduce2(s, ss);
    float inv = 1.f / (float)nelem;
    float mu = s * inv;
    float var = ss * inv - mu * mu;
    float rs = rsqrtf(var + EPSV);
    for (int e = threadIdx.x; e < nelem; e += 256) {
        int c = e / Np;
        float v = (in[ibase + e] - mu) * rs * gamma[g * cpg + c] + beta[g * cpg + c];
        out[obase + e] = v / (1.f + expf(-v));
    }
}

// ---------------- LayerNorm over D; in mode0 = (B,D,L) chan-major, mode1 = row-major
__global__ void ln_kernel(const float* __restrict__ in, float* __restrict__ out,
                          const float* __restrict__ w, const float* __restrict__ bb,
                          int D, int mode, int L, int Ctot) {
    int t = blockIdx.x;
    size_t base, stride;
    if (mode == 0) { int b = t / L; int pix = t - b * L; base = (size_t)b * Ctot * L + pix; stride = (size_t)L; }
    else { base = (size_t)t * D; stride = 1; }
    float s = 0.f, ss = 0.f;
    for (int d = threadIdx.x; d < D; d += 256) {
        float v = in[base + (size_t)d * stride]; s += v; ss += v * v;
    }
    blockReduce2(s, ss);
    float mu = s / (float)D;
    float var = ss / (float)D - mu * mu;
    float rs = rsqrtf(var + EPSV);
    for (int d = threadIdx.x; d < D; d += 256) {
        float v = in[base + (size_t)d * stride];
        out[(size_t)t * D + d] = (v - mu) * rs * w[d] + bb[d];
    }
}

// ---------------- pe = LN(pc_emb @ W_pc^T) ----------------
__global__ void pe_kernel(const float* __restrict__ pc, const float* __restrict__ Wpc,
                          const float* __restrict__ lw, const float* __restrict__ lb,
                          float* __restrict__ pe) {
    int b = blockIdx.x; int o = threadIdx.x;
    const float* pr = pc + (size_t)b * 256;
    const float* wr = Wpc + (size_t)o * 256;
    float acc = 0.f;
    for (int c = 0; c < 256; ++c) acc += pr[c] * wr[c];
    float s = acc, ss = acc * acc;
    blockReduce2(s, ss);
    float mu = s * (1.f / 256.f);
    float var = ss * (1.f / 256.f) - mu * mu;
    float rs = rsqrtf(var + EPSV);
    pe[(size_t)b * 256 + o] = (acc - mu) * rs * lw[o] + lb[o];
}

__global__ void scale_shift_kernel(const float* __restrict__ pe,
                                   const float* __restrict__ Wfs, const float* __restrict__ bfs,
                                   const float* __restrict__ Wfsh, const float* __restrict__ bfsh,
                                   float* __restrict__ scA, float* __restrict__ shA) {
    int b = blockIdx.x, o = threadIdx.x;
    const float* pr = pe + (size_t)b * 256;
    const float* w1 = Wfs + (size_t)o * 256;
    const float* w2 = Wfsh + (size_t)o * 256;
    float a = bfs[o], c = bfsh[o];
    for (int j = 0; j < 256; ++j) { float p = pr[j]; a += p * w1[j]; c += p * w2[j]; }
    scA[b * 256 + o] = a; shA[b * 256 + o] = c;
}

__global__ void gin_kernel(const float* __restrict__ pe, const float* __restrict__ xa,
                           float* __restrict__ gin) {
    int b = blockIdx.x, c = threadIdx.x;
    gin[(size_t)b * 512 + c] = pe[b * 256 + c];
    const float* xr = xa + ((size_t)b * 256 + c) * 256;
    float s = 0.f;
    for (int p = 0; p < 256; ++p) s += xr[p];
    gin[(size_t)b * 512 + 256 + c] = s * (1.f / 256.f);
}

__global__ void gate_kernel(const float* __restrict__ gin, const float* __restrict__ Wg1,
                            const float* __restrict__ bg1, const float* __restrict__ Wg2,
                            const float* __restrict__ bg2, float* __restrict__ gv) {
    int b = blockIdx.x, h = threadIdx.x;
    const float* gr = gin + (size_t)b * 512;
    const float* wr = Wg1 + (size_t)h * 512;
    float a = bg1[h];
    for (int j = 0; j < 512; ++j) a += gr[j] * wr[j];
    float hid = a / (1.f + expf(-a));
    float part = hid * Wg2[h];
    float dummy = 0.f;
    blockReduce2(part, dummy);
    if (h == 0) gv[b] = 1.f / (1.f + expf(-(part + bg2[0])));
}

// fused_in = [gated xa | positional | pe map], layout (16,768,256)
__global__ void build_fused_kernel(const float* __restrict__ xa, const float* __restrict__ scA,
                                   const float* __restrict__ shA, const float* __restrict__ gv,
                                   const float* __restrict__ Wpos, const float* __restrict__ pe,
                                   float* __restrict__ fin) {
    int idx = blockIdx.x * 256 + threadIdx.x;
    if (idx >= 16 * 768 * 256) return;
    int pix = idx & 255;
    int c = (idx >> 8) % 768;
    int b = (idx >> 8) / 768;
    float v;
    if (c < 256) {
        float xv = xa[((size_t)b * 256 + c) * 256 + pix];
        float xc = (1.f + scA[b * 256 + c]) * xv + shA[b * 256 + c];
        float g = gv[b];
        v = g * xc + (1.f - g) * xv;
    } else if (c < 512) {
        int cc = c - 256; int h = pix >> 4, w = pix & 15;
        float lx = -1.f + (2.f / 15.f) * (float)w;   // grid ch0 = xx = ls[w]
        float ly = -1.f + (2.f / 15.f) * (float)h;   // grid ch1 = yy = ls[h]
        v = Wpos[cc * 2] * lx + Wpos[cc * 2 + 1] * ly;
    } else {
        v = pe[b * 256 + (c - 512)];
    }
    fin[idx] = v;
}

__global__ void dwconv3x3_kernel(const float* __restrict__ in, const float* __restrict__ wdw,
                                 float* __restrict__ out, int C) {
    int idx = blockIdx.x * 256 + threadIdx.x;
    if (idx >= 16 * C * 256) return;
    int pix = idx & 255; int c = (idx >> 8) % C;
    int h = pix >> 4, w = pix & 15;
    const float* ib = in + ((size_t)(idx >> 8)) * 256;
    const float* wr = wdw + (size_t)c * 9;
    float s = 0.f;
    for (int kh = 0; kh < 3; ++kh) {
        int ih = h + kh - 1; if (ih < 0 || ih > 15) continue;
        for (int kw = 0; kw < 3; ++kw) {
            int iw = w + kw - 1; if (iw < 0 || iw > 15) continue;
            s += wr[kh * 3 + kw] * ib[ih * 16 + iw];
        }
    }
    out[idx] = s;
}

__global__ void mean_bcast_kernel(const float* __restrict__ o0, float* __restrict__ cat) {
    int b = blockIdx.x, c = threadIdx.x;
    const float* r = o0 + ((size_t)b * 256 + c) * 256;
    float s = 0.f;
    for (int p = 0; p < 256; ++p) s += r[p];
    float m = s * (1.f / 256.f);
    float* ob = cat + ((size_t)b * 1280 + 256 + c) * 256;
    for (int p = 0; p < 256; ++p) ob[p] = m;
}

// adaptive pool to s x s, output token-major (b*ss+pix, 768)
__global__ void pool_kernel(const float* __restrict__ fin, float* __restrict__ poolT, int s) {
    int idx = blockIdx.x * 256 + threadIdx.x;
    int ss = s * s;
    if (idx >= 16 * 768 * ss) return;
    int p = idx % s; int o = (idx / s) % s; int c = (idx / ss) % 768; int b = idx / (768 * ss);
    int sh = (o * 16) / s, eh = ((o + 1) * 16 + s - 1) / s;
    int sw = (p * 16) / s, ew = ((p + 1) * 16 + s - 1) / s;
    const float* ib = fin + ((size_t)b * 768 + c) * 256;
    float acc = 0.f;
    for (int ih = sh; ih < eh; ++ih)
        for (int iw = sw; iw < ew; ++iw) acc += ib[ih * 16 + iw];
    acc /= (float)((eh - sh) * (ew - sw));
    poolT[((size_t)b * ss + o * s + p) * 768 + c] = acc;
}

// bilinear upsample s x s -> 16 x 16 into cat channel slice c0
__global__ void upsample_kernel(const float* __restrict__ pr, float* __restrict__ cat,
                                int s, int c0) {
    int idx = blockIdx.x * 256 + threadIdx.x;
    if (idx >= 16 * 256 * 256) return;
    int w = idx & 15, h = (idx >> 4) & 15, c = (idx >> 8) & 255, b = idx >> 16;
    float fs = (float)s;
    float shf = (h + 0.5f) * fs / 16.f - 0.5f; shf = fminf(fmaxf(shf, 0.f), fs - 1.f);
    float swf = (w + 0.5f) * fs / 16.f - 0.5f; swf = fminf(fmaxf(swf, 0.f), fs - 1.f);
    int lh = (int)shf; int hh = min(lh + 1, s - 1); float fh = shf - (float)lh;
    int lw = (int)swf; int hw = min(lw + 1, s - 1); float fw = swf - (float)lw;
    const float* p = pr + ((size_t)b * 256 + c) * (size_t)(s * s);
    float v = (1.f - fh) * ((1.f - fw) * p[lh * s + lw] + fw * p[lh * s + hw])
            +        fh  * ((1.f - fw) * p[hh * s + lw] + fw * p[hh * s + hw]);
    cat[((size_t)b * 1280 + c0 + c) * 256 + (h * 16 + w)] = v;
}

// causal depthwise conv1d (K=4) + SiLU over x-half of xz
__global__ void conv1d_silu_kernel(const float* __restrict__ xz, const float* __restrict__ cw,
                                   const float* __restrict__ cb, float* __restrict__ u) {
    size_t idx = (size_t)blockIdx.x * 256 + threadIdx.x;
    if (idx >= (size_t)4096 * 2560) return;
    int d = (int)(idx % 2560);
    int t = (int)(idx / 2560);
    int l = t & 255;
    float s = cb[d];
    const float* cwd = cw + (size_t)d * 4;
    for (int k = 0; k < 4; ++k) {
        int lp = l - 3 + k;
        if (lp >= 0) s += cwd[k] * xz[(size_t)(t - 3 + k) * 5120 + d];
    }
    u[idx] = s / (1.f + expf(-s));
}

// selective scan: one lane per (b,d), 16 states in registers; B/C staged via LDS
__global__ void scan_kernel(const float* __restrict__ dt, const float* __restrict__ u,
                            const float* __restrict__ xdbl, const float* __restrict__ xz,
                            const float* __restrict__ Alog, const float* __restrict__ Dp,
                            float* __restrict__ y) {
    int b = blockIdx.y;
    int d = blockIdx.x * 256 + threadIdx.x;
    float A[16], h[16];
#pragma unroll
    for (int n = 0; n < 16; ++n) { A[n] = -expf(Alog[(size_t)d * 16 + n]); h[n] = 0.f; }
    float Dv = Dp[d];
    __shared__ float sBC[32];
    for (int l = 0; l < 256; ++l) {
        size_t t = (size_t)b * 256 + l;
        if (threadIdx.x < 32) sBC[threadIdx.x] = xdbl[t * 112 + 80 + threadIdx.x];
        __syncthreads();
        float dtv = dt[t * 2560 + d];
        float uv  = u[t * 2560 + d];
        float du = dtv * uv;
        float yv = 0.f;
#pragma unroll
        for (int n = 0; n < 16; ++n) {
            float dA = expf(dtv * A[n]);
            h[n] = dA * h[n] + du * sBC[n];
            yv += h[n] * sBC[16 + n];
        }
        yv += uv * Dv;
        float zv = xz[t * 5120 + 2560 + d];
        yv *= zv / (1.f + expf(-zv));
        __syncthreads();
        y[t * 2560 + d] = yv;    // y may alias dt: same (t,d) read first above
    }
}

// ---------------- host side ----------------
static inline void launch_gemm(hipStream_t s, const float* A, const float* W, float* C,
                               const float* bias, int M, int N, int K, int lda,
                               int amode, int aL, int cmode, int cL, int cCtot, int act) {
    dim3 g((N + 63) / 64, (M + 127) / 128), b(256);
    wmma_gemm_f32<<<g, b, 0, s>>>(A, W, C, bias, M, N, K, lda, amode, aL, cmode, cL, cCtot, act);
}

extern "C" void kernel_launch(void* const* d_in, const int* in_sizes, int n_in,
                              void* d_out, int out_size, void* d_ws, size_t ws_size,
                              hipStream_t stream) {
    const float* x       = (const float*)d_in[0];
    const float* pc      = (const float*)d_in[1];
    const float* W_ia    = (const float*)d_in[2];
    const float* gn_ia_w = (const float*)d_in[3];
    const float* gn_ia_b = (const float*)d_in[4];
    const float* W_pc    = (const float*)d_in[5];
    const float* ln_pc_w = (const float*)d_in[6];
    const float* ln_pc_b = (const float*)d_in[7];
    const float* W_pos   = (const float*)d_in[8];
    const float* W_fs    = (const float*)d_in[9];
    const float* b_fs    = (const float*)d_in[10];
    const float* W_fsh   = (const float*)d_in[11];
    const float* b_fsh   = (const float*)d_in[12];
    const float* W_g1    = (const float*)d_in[13];
    const float* b_g1    = (const float*)d_in[14];
    const float* W_g2    = (const float*)d_in[15];
    const float* b_g2    = (const float*)d_in[16];
    const float* dw1     = (const float*)d_in[17];
    const float* pw1     = (const float*)d_in[18];
    const float* gnb1_w  = (const float*)d_in[19];
    const float* gnb1_b  = (const float*)d_in[20];
    const float* dw2     = (const float*)d_in[21];
    const float* pw2     = (const float*)d_in[22];
    const float* gnb2_w  = (const float*)d_in[23];
    const float* gnb2_b  = (const float*)d_in[24];
    const float* pre_w   = (const float*)d_in[25];
    const float* pre_b   = (const float*)d_in[26];
    const float* post_w  = (const float*)d_in[27];
    const float* post_b  = (const float*)d_in[28];
    const float* W_in    = (const float*)d_in[29];
    const float* conv_w  = (const float*)d_in[30];
    const float* conv_b  = (const float*)d_in[31];
    const float* W_xproj = (const float*)d_in[32];
    const float* W_dt    = (const float*)d_in[33];
    const float* b_dt    = (const float*)d_in[34];
    const float* A_log   = (const float*)d_in[35];
    const float* Dp      = (const float*)d_in[36];
    const float* W_out   = (const float*)d_in[37];
    const float* W_op    = (const float*)d_in[38];
    const float* gn_op_w = (const float*)d_in[39];
    const float* gn_op_b = (const float*)d_in[40];
    const float* W_p[4]; const float* gnp_w[4]; const float* gnp_b[4];
    for (int i = 0; i < 4; ++i) {
        W_p[i]   = (const float*)d_in[41 + 3 * i];
        gnp_w[i] = (const float*)d_in[42 + 3 * i];
        gnp_b[i] = (const float*)d_in[43 + 3 * i];
    }

    float* ws = (float*)d_ws;
    size_t o = 0;
    float* xaB   = ws + o; o += (size_t)16 * 256 * 256;   // 1.05M
    float* finB  = ws + o; o += (size_t)16 * 768 * 256;   // 3.15M
    float* t768  = ws + o; o += (size_t)16 * 768 * 256;   // 3.15M scratch
    float* f256  = ws + o; o += (size_t)16 * 256 * 256;
    float* poolT = ws + o; o += (size_t)2704 * 768;
    float* praw  = ws + o; o += (size_t)16 * 256 * 169;
    float* catB  = ws + o; o += (size_t)4096 * 1280;      // cat, later tok2
    float* tokN  = ws + o; o += (size_t)4096 * 1280;      // pre-LN tokens, later post-LN
    float* xzB   = ws + o; o += (size_t)4096 * 5120;
    float* uB    = ws + o; o += (size_t)4096 * 2560;
    float* xdblB = ws + o; o += (size_t)4096 * 112;
    float* dtB   = ws + o; o += (size_t)4096 * 2560;      // dt, then y (aliased)
    float* peB   = ws + o; o += 4096;
    float* scA   = ws + o; o += 4096;
    float* shA   = ws + o; o += 4096;
    float* ginB  = ws + o; o += 8192;
    float* gvB   = ws + o; o += 16;
    (void)ws_size; (void)in_sizes; (void)n_in; (void)out_size;

    // pe / film params / gate inputs
    pe_kernel<<<16, 256, 0, stream>>>(pc, W_pc, ln_pc_w, ln_pc_b, peB);
    scale_shift_kernel<<<16, 256, 0, stream>>>(peB, W_fs, b_fs, W_fsh, b_fsh, scA, shA);

    // xa = silu(GN(conv1x1(x, W_ia)))
    launch_gemm(stream, x, W_ia, xaB, nullptr, 4096, 256, 512, 0, 1, 256, 1, 256, 256, 0);
    gn_silu_kernel<<<dim3(32, 16), 256, 0, stream>>>(xaB, xaB, gn_ia_w, gn_ia_b, 256, 256, 256, 0, 256, 0);

    gin_kernel<<<16, 256, 0, stream>>>(peB, xaB, ginB);
    gate_kernel<<<16, 256, 0, stream>>>(ginB, W_g1, b_g1, W_g2, b_g2, gvB);
    build_fused_kernel<<<(16 * 768 * 256) / 256, 256, 0, stream>>>(xaB, scA, shA, gvB, W_pos, peB, finB);

    // dsconv1: dw3x3 -> pw1 -> GN -> silu
    dwconv3x3_kernel<<<(16 * 768 * 256) / 256, 256, 0, stream>>>(finB, dw1, t768, 768);
    launch_gemm(stream, t768, pw1, f256, nullptr, 4096, 256, 768, 0, 1, 256, 1, 256, 256, 0);
    gn_silu_kernel<<<dim3(32, 16), 256, 0, stream>>>(f256, f256, gnb1_w, gnb1_b, 256, 256, 256, 0, 256, 0);
    // dsconv2 -> cat channels [0,256)
    dwconv3x3_kernel<<<(16 * 256 * 256) / 256, 256, 0, stream>>>(f256, dw2, t768, 256);
    launch_gemm(stream, t768, pw2, catB, nullptr, 4096, 256, 256, 0, 1, 256, 1, 256, 1280, 0);
    gn_silu_kernel<<<dim3(32, 16), 256, 0, stream>>>(catB, catB, gnb2_w, gnb2_b, 256, 256, 1280, 0, 1280, 0);

    // PPM branch 0: global mean of silu(GN(conv(fused_in, W_p0))) -> cat [256,512)
    launch_gemm(stream, finB, W_p[0], t768, nullptr, 4096, 256, 768, 0, 1, 256, 1, 256, 256, 0);
    gn_silu_kernel<<<dim3(32, 16), 256, 0, stream>>>(t768, t768, gnp_w[0], gnp_b[0], 256, 256, 256, 0, 256, 0);
    mean_bcast_kernel<<<16, 256, 0, stream>>>(t768, catB);

    // PPM pooled branches s = 5, 9, 13 -> cat [512,768),[768,1024),[1024,1280)
    int psz[3] = {5, 9, 13};
    for (int i = 0; i < 3; ++i) {
        int s = psz[i]; int ss = s * s; int M = 16 * ss;
        int tot = 16 * 768 * ss;
        pool_kernel<<<(tot + 255) / 256, 256, 0, stream>>>(finB, poolT, s);
        launch_gemm(stream, poolT, W_p[i + 1], praw, nullptr, M, 256, 768, 768, 0, 0, 1, ss, 256, 0);
        gn_silu_kernel<<<dim3(32, 16), 256, 0, stream>>>(praw, praw, gnp_w[i + 1], gnp_b[i + 1], 256, ss, 256, 0, 256, 0);
        upsample_kernel<<<(16 * 256 * 256) / 256, 256, 0, stream>>>(praw, catB, s, 256 * (i + 2));
    }

    // tokens = preLN(cat)  (chan-major -> token-major)
    ln_kernel<<<4096, 256, 0, stream>>>(catB, tokN, pre_w, pre_b, 1280, 0, 256, 1280);

    // Mamba block
    launch_gemm(stream, tokN, W_in, xzB, nullptr, 4096, 5120, 1280, 1280, 0, 0, 0, 0, 0, 0);
    conv1d_silu_kernel<<<(4096 * 2560) / 256, 256, 0, stream>>>(xzB, conv_w, conv_b, uB);
    launch_gemm(stream, uB, W_xproj, xdblB, nullptr, 4096, 112, 2560, 2560, 0, 0, 0, 0, 0, 0);
    launch_gemm(stream, xdblB, W_dt, dtB, b_dt, 4096, 2560, 80, 112, 0, 0, 0, 0, 0, 1);
    scan_kernel<<<dim3(10, 16), 256, 0, stream>>>(dtB, uB, xdblB, xzB, A_log, Dp, dtB);
    launch_gemm(stream, dtB, W_out, catB, nullptr, 4096, 1280, 2560, 2560, 0, 0, 0, 0, 0, 0);

    // postLN -> final conv1x1 -> GN+SiLU -> d_out (16,128,16,16)
    ln_kernel<<<4096, 256, 0, stream>>>(catB, tokN, post_w, post_b, 1280, 1, 0, 0);
    launch_gemm(stream, tokN, W_op, t768, nullptr, 4096, 128, 1280, 1280, 0, 0, 1, 256, 128, 0);
    gn_silu_kernel<<<dim3(32, 16), 256, 0, stream>>>(t768, (float*)d_out, gn_op_w, gn_op_b,
                                                     128, 256, 128, 0, 128, 0);
}